// ImprovedGNN_15247133901708
// MI455X (gfx1250) — compile-verified
//
#include <hip/hip_runtime.h>
#include <hip/hip_bf16.h>
#include <math.h>

typedef __attribute__((ext_vector_type(2))) float v2f;
typedef __attribute__((ext_vector_type(8))) float v8f;

#define Hdim 128
#define Ddim 64

// ---------------------------------------------------------------------------
// Degree / normalization
// ---------------------------------------------------------------------------
__global__ void degree_kernel(const int* __restrict__ dst, float* __restrict__ deg, int E) {
    int e = blockIdx.x * blockDim.x + threadIdx.x;
    if (e < E) atomicAdd(&deg[dst[e]], 1.0f);
}

__global__ void dinv_kernel(float* __restrict__ deg, int N) {
    int i = blockIdx.x * blockDim.x + threadIdx.x;
    if (i < N) deg[i] = rsqrtf(deg[i] + 1.0f);   // +1 for self-loop; always > 0
}

// ---------------------------------------------------------------------------
// WMMA fp32 GEMM:  Hout[N,128] = X[N,128] @ W[128,128]
// One 256-thread block (8 waves) per 16-row tile; wave w owns cols [16w,16w+16).
// Uses V_WMMA_F32_16X16X4_F32, K accumulated in 32 steps of 4.
// ---------------------------------------------------------------------------
__global__ void gemm_wmma_kernel(const float* __restrict__ X,
                                 const float* __restrict__ W,
                                 float* __restrict__ Hout, int N) {
    const int wave  = threadIdx.x >> 5;
    const int lane  = threadIdx.x & 31;
    const int row16 = blockIdx.x * 16;
    const int col16 = wave * 16;
    if (row16 >= N) return;

    const int mn   = lane & 15;            // M index for A-frag, N index for B-frag
    const int soff = (lane >> 4) * 2;      // lanes 16-31 hold K sub-offset +2

    const float* xrow = X + (size_t)(row16 + mn) * Hdim;

    v8f acc = {};
#pragma unroll
    for (int k0 = 0; k0 < Hdim; k0 += 4) {
        v2f a, b;
        // A 16x4 fp32 frag: lane m<16 -> K=k0..k0+1 ; lane m+16 -> K=k0+2..k0+3
        a.x = xrow[k0 + soff];
        a.y = xrow[k0 + soff + 1];
        // B 4x16 fp32 frag (W row-major, W[k][n]): same K sub-offset pattern
        b.x = W[(size_t)(k0 + soff)     * Hdim + col16 + mn];
        b.y = W[(size_t)(k0 + soff + 1) * Hdim + col16 + mn];
        acc = __builtin_amdgcn_wmma_f32_16x16x4_f32(
            /*neg_a=*/false, a, /*neg_b=*/false, b,
            /*c_mod=*/(short)0, acc, /*reuse_a=*/false, /*reuse_b=*/false);
    }

    // C/D layout: VGPR r -> M=r (lanes 0-15), M=8+r (lanes 16-31); N = lane&15
    const int mh = (lane >> 4) * 8;
#pragma unroll
    for (int r = 0; r < 8; ++r) {
        Hout[(size_t)(row16 + mh + r) * Hdim + col16 + mn] = acc[r];
    }
}

// ---------------------------------------------------------------------------
// Edge scatter: out[dst] += h[src] * dinv[src]*dinv[dst]   (one wave per edge)
// ---------------------------------------------------------------------------
__global__ void agg_kernel(const float* __restrict__ h,
                           const int* __restrict__ src,
                           const int* __restrict__ dst,
                           const float* __restrict__ dinv,
                           float* __restrict__ out, int E) {
    int wid  = blockIdx.x * (blockDim.x >> 5) + (threadIdx.x >> 5);
    int lane = threadIdx.x & 31;
    if (wid >= E) return;
    int s = src[wid];
    int d = dst[wid];
    float w = dinv[s] * dinv[d];
    const float4* hs = (const float4*)(h + (size_t)s * Hdim);
    // hint the gathered row into cache (global_prefetch_b8 on gfx1250)
    __builtin_prefetch((const void*)hs, 0, 0);
    float4 v = hs[lane];
    float* o = out + (size_t)d * Hdim + lane * 4;
    atomicAdd(o + 0, v.x * w);
    atomicAdd(o + 1, v.y * w);
    atomicAdd(o + 2, v.z * w);
    atomicAdd(o + 3, v.w * w);
}

// ---------------------------------------------------------------------------
// Fused epilogue: self-loop + bias + BN(eval) + LeakyReLU + optional residual
// ---------------------------------------------------------------------------
__global__ void epilogue_kernel(const float* __restrict__ agg,
                                const float* __restrict__ h,
                                const float* __restrict__ dinv,
                                const float* __restrict__ bias,
                                const float* __restrict__ gma,
                                const float* __restrict__ bet,
                                const float* __restrict__ mu,
                                const float* __restrict__ var,
                                const float* __restrict__ res,   // may be null
                                float* __restrict__ out, int N) {
    long i = (long)blockIdx.x * blockDim.x + threadIdx.x;
    long total = (long)N * Hdim;
    if (i >= total) return;
    int  c    = (int)(i & (Hdim - 1));
    int  node = (int)(i >> 7);
    float di  = dinv[node];
    float val = agg[i] + h[i] * di * di + bias[c];
    float t   = (val - mu[c]) * rsqrtf(var[c] + 1e-5f) * gma[c] + bet[c];
    t = (t >= 0.0f) ? t : 0.01f * t;
    if (res) t += res[i];
    out[i] = t;
}

// ---------------------------------------------------------------------------
// Global mean pool (atomic accumulate), one block (128 threads) per node
// ---------------------------------------------------------------------------
__global__ void pool_kernel(const float* __restrict__ x3,
                            const int* __restrict__ batch,
                            float* __restrict__ pooled,
                            float* __restrict__ cnt, int N) {
    int i = blockIdx.x;
    if (i >= N) return;
    int g = batch[i];
    int t = threadIdx.x;
    atomicAdd(&pooled[(size_t)g * Hdim + t], x3[(size_t)i * Hdim + t]);
    if (t == 0) atomicAdd(&cnt[g], 1.0f);
}

// ---------------------------------------------------------------------------
// MLP head + L2 normalize, one block (128 threads) per graph
// ---------------------------------------------------------------------------
__global__ void mlp_kernel(const float* __restrict__ pooled,
                           const float* __restrict__ cnt,
                           const float* __restrict__ fcW1,
                           const float* __restrict__ fcb1,
                           const float* __restrict__ fcW2,
                           const float* __restrict__ fcb2,
                           float* __restrict__ out) {
    __shared__ float p[Hdim];
    __shared__ float hb[Hdim];
    __shared__ float ob[Ddim];
    __shared__ float nrm;
    int g = blockIdx.x;
    int t = threadIdx.x;

    float c = fmaxf(cnt[g], 1.0f);
    p[t] = pooled[(size_t)g * Hdim + t] / c;
    __syncthreads();

    float acc = fcb1[t];
#pragma unroll 8
    for (int k = 0; k < Hdim; ++k) acc += p[k] * fcW1[(size_t)k * Hdim + t];
    hb[t] = (acc >= 0.0f) ? acc : 0.01f * acc;
    __syncthreads();

    if (t < Ddim) {
        float o2 = fcb2[t];
#pragma unroll 8
        for (int k = 0; k < Hdim; ++k) o2 += hb[k] * fcW2[(size_t)k * Ddim + t];
        ob[t] = o2;
    }
    __syncthreads();

    if (t == 0) {
        float s = 0.0f;
        for (int k = 0; k < Ddim; ++k) s += ob[k] * ob[k];
        nrm = fmaxf(sqrtf(s), 1e-12f);
    }
    __syncthreads();

    if (t < Ddim) out[(size_t)g * Ddim + t] = ob[t] / nrm;
}

// ---------------------------------------------------------------------------
// Launch
// ---------------------------------------------------------------------------
extern "C" void kernel_launch(void* const* d_in, const int* in_sizes, int n_in,
                              void* d_out, int out_size, void* d_ws, size_t ws_size,
                              hipStream_t stream) {
    const float* x    = (const float*)d_in[0];
    const int*   ei   = (const int*)  d_in[1];
    const int*   batc = (const int*)  d_in[2];
    const float* W1   = (const float*)d_in[3];
    const float* b1   = (const float*)d_in[4];
    const float* W2   = (const float*)d_in[5];
    const float* b2   = (const float*)d_in[6];
    const float* W3   = (const float*)d_in[7];
    const float* b3   = (const float*)d_in[8];
    const float* g1   = (const float*)d_in[9];
    const float* be1  = (const float*)d_in[10];
    const float* m1   = (const float*)d_in[11];
    const float* v1   = (const float*)d_in[12];
    const float* g2   = (const float*)d_in[13];
    const float* be2  = (const float*)d_in[14];
    const float* m2   = (const float*)d_in[15];
    const float* v2   = (const float*)d_in[16];
    const float* g3   = (const float*)d_in[17];
    const float* be3  = (const float*)d_in[18];
    const float* m3   = (const float*)d_in[19];
    const float* v3   = (const float*)d_in[20];
    const float* fcW1 = (const float*)d_in[21];
    const float* fcb1 = (const float*)d_in[22];
    const float* fcW2 = (const float*)d_in[23];
    const float* fcb2 = (const float*)d_in[24];

    const int N = in_sizes[2];          // batch is [N]
    const int E = in_sizes[1] / 2;      // edge_index is [2,E]
    const int G = 64;

    const int* src = ei;
    const int* dst = ei + E;

    // workspace carve-up
    const size_t NH = (size_t)N * Hdim;
    char* ws = (char*)d_ws;
    float* buf0   = (float*)ws;                     ws += NH * sizeof(float);   // h (pre-agg)
    float* buf1   = (float*)ws;                     ws += NH * sizeof(float);   // agg / x2
    float* buf2   = (float*)ws;                     ws += NH * sizeof(float);   // x1 / agg3 / x3
    float* dinv   = (float*)ws;                     ws += (size_t)N * sizeof(float);
    float* pooled = (float*)ws;                     ws += (size_t)G * Hdim * sizeof(float);
    float* cnt    = (float*)ws;                     ws += (size_t)G * sizeof(float);

    const int TPB       = 256;
    const int gemmGrid  = (N + 15) / 16;
    const int aggGrid   = (E + 7) / 8;              // 8 waves (edges) per block
    const int epiGrid   = (int)((NH + TPB - 1) / TPB);

    // degree -> dinv (in place)
    hipMemsetAsync(dinv, 0, (size_t)N * sizeof(float), stream);
    degree_kernel<<<(E + TPB - 1) / TPB, TPB, 0, stream>>>(dst, dinv, E);
    dinv_kernel<<<(N + TPB - 1) / TPB, TPB, 0, stream>>>(dinv, N);

    // ---- layer 1 : x1 = lrelu(bn(agg(x@W1)+b1)) -> buf2
    gemm_wmma_kernel<<<gemmGrid, TPB, 0, stream>>>(x, W1, buf0, N);
    hipMemsetAsync(buf1, 0, NH * sizeof(float), stream);
    agg_kernel<<<aggGrid, TPB, 0, stream>>>(buf0, src, dst, dinv, buf1, E);
    epilogue_kernel<<<epiGrid, TPB, 0, stream>>>(buf1, buf0, dinv, b1,
                                                 g1, be1, m1, v1, nullptr, buf2, N);

    // ---- layer 2 : x2 = lrelu(bn(agg(x1@W2)+b2)) + x1 -> buf1 (in place over agg)
    gemm_wmma_kernel<<<gemmGrid, TPB, 0, stream>>>(buf2, W2, buf0, N);
    hipMemsetAsync(buf1, 0, NH * sizeof(float), stream);
    agg_kernel<<<aggGrid, TPB, 0, stream>>>(buf0, src, dst, dinv, buf1, E);
    epilogue_kernel<<<epiGrid, TPB, 0, stream>>>(buf1, buf0, dinv, b2,
                                                 g2, be2, m2, v2, buf2, buf1, N);

    // ---- layer 3 : x3 = lrelu(bn(agg(x2@W3)+b3)) + x2 -> buf2 (in place over agg)
    gemm_wmma_kernel<<<gemmGrid, TPB, 0, stream>>>(buf1, W3, buf0, N);
    hipMemsetAsync(buf2, 0, NH * sizeof(float), stream);
    agg_kernel<<<aggGrid, TPB, 0, stream>>>(buf0, src, dst, dinv, buf2, E);
    epilogue_kernel<<<epiGrid, TPB, 0, stream>>>(buf2, buf0, dinv, b3,
                                                 g3, be3, m3, v3, buf1, buf2, N);

    // ---- mean pool + MLP head + L2 normalize
    hipMemsetAsync(pooled, 0, (size_t)G * Hdim * sizeof(float), stream);
    hipMemsetAsync(cnt, 0, (size_t)G * sizeof(float), stream);
    pool_kernel<<<N, Hdim, 0, stream>>>(buf2, batc, pooled, cnt, N);
    mlp_kernel<<<G, Hdim, 0, stream>>>(pooled, cnt, fcW1, fcb1, fcW2, fcb2,
                                       (float*)d_out);
}